// gtransformerlayer_19859928777156
// MI455X (gfx1250) — compile-verified
//
#include <hip/hip_runtime.h>
#include <hip/hip_bf16.h>

typedef __attribute__((ext_vector_type(2))) float v2f;
typedef __attribute__((ext_vector_type(8))) float v8f;

#define ALPHA_C     0.15f
#define NEG_SLOPE_C 0.2f

// ---------------------------------------------------------------------------
// WMMA fp32 GEMM:  D[M,N] = epilogue( A[M,K] @ B[N,K]^T )   (torch Linear)
// Each wave computes a 16x64 tile of C: one A fragment feeds 4 WMMAs.
// Block = 256 threads = 8 waves -> 128x64 block tile.
// MODE 0: D = C
// MODE 1: D = res + scale*C                      (out-proj rezero residual)
// MODE 2: D = relu(C + bias)                     (FFN layer 1)
// MODE 3: D = res + scale*(C + bias)             (FFN layer 2 rezero residual)
// ---------------------------------------------------------------------------
__device__ __forceinline__ v2f ld2f(const float* p) { return *(const v2f*)p; }

template <int MODE>
__global__ __launch_bounds__(256) void gemm_wmma_f32_k(
    const float* __restrict__ A, const float* __restrict__ B,
    float* __restrict__ D, const float* __restrict__ bias,
    const float* __restrict__ res, const float* __restrict__ scale,
    int M, int N, int K)
{
    const int lane = threadIdx.x & 31;
    const int wave = threadIdx.x >> 5;
    const int m0 = blockIdx.x * 128 + wave * 16;
    const int n0 = blockIdx.y * 64;
    if (m0 >= M) return;                 // wave-uniform exit (EXEC stays all-1s)
    const int half = lane >> 4;          // 0: K=0..1 , 1: K=2..3
    const int l16  = lane & 15;

    int arow = m0 + l16; if (arow >= M) arow = M - 1;   // clamp partial M tile
    const float* Ap  = A + (size_t)arow * K + 2 * half;
    const float* Bp0 = B + (size_t)(n0 + l16) * K + 2 * half;
    const float* Bp1 = Bp0 + (size_t)16 * K;
    const float* Bp2 = Bp0 + (size_t)32 * K;
    const float* Bp3 = Bp0 + (size_t)48 * K;

    v8f c0 = {}, c1 = {}, c2 = {}, c3 = {};
#pragma unroll 2
    for (int k = 0; k < K; k += 4) {
        v2f a  = ld2f(Ap  + k);
        v2f b0 = ld2f(Bp0 + k);
        v2f b1 = ld2f(Bp1 + k);
        v2f b2 = ld2f(Bp2 + k);
        v2f b3 = ld2f(Bp3 + k);
        c0 = __builtin_amdgcn_wmma_f32_16x16x4_f32(false, a, false, b0, (short)0, c0, false, false);
        c1 = __builtin_amdgcn_wmma_f32_16x16x4_f32(false, a, false, b1, (short)0, c1, false, false);
        c2 = __builtin_amdgcn_wmma_f32_16x16x4_f32(false, a, false, b2, (short)0, c2, false, false);
        c3 = __builtin_amdgcn_wmma_f32_16x16x4_f32(false, a, false, b3, (short)0, c3, false, false);
    }

    const float s = (MODE == 1 || MODE == 3) ? scale[0] : 1.0f;
    auto store_tile = [&](v8f c, int ncol0) {
#pragma unroll
        for (int j = 0; j < 8; ++j) {
            int row = m0 + j + 8 * half;
            if (row < M) {
                int col = ncol0 + l16;
                float v = c[j];
                if (MODE == 2 || MODE == 3) v += bias[col];
                if (MODE == 2) v = fmaxf(v, 0.0f);
                size_t idx = (size_t)row * N + col;
                if (MODE == 1 || MODE == 3) v = res[idx] + s * v;
                D[idx] = v;
            }
        }
    };
    store_tile(c0, n0);
    store_tile(c1, n0 + 16);
    store_tile(c2, n0 + 32);
    store_tile(c3, n0 + 48);
}

// ---------------------------------------------------------------------------
// Elementwise / graph kernels
// ---------------------------------------------------------------------------
__global__ void fill_kernel(float* __restrict__ p, float v, long n)
{
    long i = (long)blockIdx.x * blockDim.x + threadIdx.x;
    long stride = (long)gridDim.x * blockDim.x;
    for (; i < n; i += stride) p[i] = v;
}

// eh[n,h] = sum_d feat[n, h*32+d] * attn[h*32+d]
__global__ void logits_kernel(const float* __restrict__ feat,
                              const float* __restrict__ attn,
                              float* __restrict__ out, int N)
{
    int idx = blockIdx.x * blockDim.x + threadIdx.x;
    if (idx >= N * 8) return;
    int n = idx >> 3, h = idx & 7;
    const float* f = feat + (size_t)n * 256 + h * 32;
    const float* w = attn + h * 32;
    float s = 0.f;
#pragma unroll
    for (int d = 0; d < 32; ++d) s += f[d] * w[d];
    out[idx] = s;
}

__device__ __forceinline__ void atomicMaxF(float* addr, float val)
{
    int old = __float_as_int(*addr);
    while (__int_as_float(old) < val) {
        int assumed = old;
        old = atomicCAS((int*)addr, assumed, __float_as_int(val));
        if (old == assumed) break;
    }
}

__global__ void edge_score_kernel(const int* __restrict__ src,
                                  const int* __restrict__ dst,
                                  const int* __restrict__ lab,
                                  const float* __restrict__ eh,
                                  const float* __restrict__ et,
                                  const float* __restrict__ er,
                                  float* __restrict__ escore,
                                  float* __restrict__ m, int E)
{
    int idx = blockIdx.x * blockDim.x + threadIdx.x;
    if (idx >= E * 8) return;
    int e = idx >> 3, h = idx & 7;
    float v = eh[src[e] * 8 + h] + et[dst[e] * 8 + h] + er[lab[e] * 8 + h];
    v = (v > 0.f) ? v : NEG_SLOPE_C * v;
    escore[idx] = v;
    atomicMaxF(&m[dst[e] * 8 + h], v);
}

__global__ void edge_exp_kernel(const int* __restrict__ dst,
                                float* __restrict__ escore,
                                const float* __restrict__ m,
                                float* __restrict__ z, int E)
{
    int idx = blockIdx.x * blockDim.x + threadIdx.x;
    if (idx >= E * 8) return;
    int e = idx >> 3, h = idx & 7;
    int d8 = dst[e] * 8 + h;
    float v = expf(escore[idx] - m[d8]);
    escore[idx] = v;
    atomicAdd(&z[d8], v);
}

__global__ void edge_norm_kernel(const int* __restrict__ dst,
                                 float* __restrict__ escore,
                                 const float* __restrict__ z, int E)
{
    int idx = blockIdx.x * blockDim.x + threadIdx.x;
    if (idx >= E * 8) return;
    int e = idx >> 3, h = idx & 7;
    escore[idx] = escore[idx] / z[dst[e] * 8 + h];
}

// One edge per 64 lanes; each lane handles a float4 (h = lane/8, d = lane%8*4)
__global__ __launch_bounds__(256) void hop_kernel(
    const int* __restrict__ src, const int* __restrict__ dst,
    const float* __restrict__ a, const float* __restrict__ fin,
    float* __restrict__ acc, int E)
{
    int e = blockIdx.x * 4 + (threadIdx.x >> 6);
    if (e >= E) return;
    int t = threadIdx.x & 63;
    int s = src[e], d = dst[e];
    float av = a[(size_t)e * 8 + (t >> 3)];
    float4 f = ((const float4*)(fin + (size_t)s * 256))[t];
    float* out = acc + (size_t)d * 256 + t * 4;
    atomicAdd(out + 0, av * f.x);
    atomicAdd(out + 1, av * f.y);
    atomicAdd(out + 2, av * f.z);
    atomicAdd(out + 3, av * f.w);
}

__global__ void combine_kernel(const float4* __restrict__ acc,
                               const float4* __restrict__ base,
                               float4* __restrict__ out, long n4)
{
    long i = (long)blockIdx.x * blockDim.x + threadIdx.x;
    long stride = (long)gridDim.x * blockDim.x;
    for (; i < n4; i += stride) {
        float4 a = acc[i], b = base[i], r;
        r.x = (1.f - ALPHA_C) * a.x + ALPHA_C * b.x;
        r.y = (1.f - ALPHA_C) * a.y + ALPHA_C * b.y;
        r.z = (1.f - ALPHA_C) * a.z + ALPHA_C * b.z;
        r.w = (1.f - ALPHA_C) * a.w + ALPHA_C * b.w;
        out[i] = r;
    }
}

__global__ void relrst_kernel(const float* __restrict__ rel_embed,
                              const float* __restrict__ rel_feat,
                              const float* __restrict__ rw,
                              float* __restrict__ out, int n)
{
    int i = blockIdx.x * blockDim.x + threadIdx.x;
    if (i < n) out[i] = rel_embed[i] + rw[0] * rel_feat[i];
}

// ---------------------------------------------------------------------------
// Host launch sequence
// ---------------------------------------------------------------------------
extern "C" void kernel_launch(void* const* d_in, const int* in_sizes, int n_in,
                              void* d_out, int out_size, void* d_ws, size_t ws_size,
                              hipStream_t stream)
{
    (void)in_sizes; (void)n_in; (void)out_size; (void)ws_size;

    const float* ent_embed = (const float*)d_in[0];
    const float* rel_embed = (const float*)d_in[1];
    const int*   src       = (const int*)d_in[2];
    const int*   dst       = (const int*)d_in[3];
    const int*   lab       = (const int*)d_in[4];
    const float* W_ent     = (const float*)d_in[5];
    const float* W_rel     = (const float*)d_in[6];
    const float* W_out     = (const float*)d_in[7];
    const float* attn_h    = (const float*)d_in[8];
    const float* attn_t    = (const float*)d_in[9];
    const float* attn_r    = (const float*)d_in[10];
    const float* w1        = (const float*)d_in[11];
    const float* b1        = (const float*)d_in[12];
    const float* w2        = (const float*)d_in[13];
    const float* b2        = (const float*)d_in[14];
    const float* rw1       = (const float*)d_in[15];
    const float* rw2       = (const float*)d_in[16];
    const float* rw_rel    = (const float*)d_in[17];

    const int NENT = 50000, NREL = 500, E = 800000, DM = 256, HH = 8, DFF = 1024;

    char* ws = (char*)d_ws;
    size_t off = 0;
    auto take = [&](size_t nf) -> float* {
        float* p = (float*)(ws + off);
        off += ((nf * sizeof(float) + 255) & ~(size_t)255);
        return p;
    };
    float* ent_feat = take((size_t)NENT * DM);
    float* feat_cur = take((size_t)NENT * DM);
    float* acc      = take((size_t)NENT * DM);
    float* ent_rst  = take((size_t)NENT * DM);
    float* hid      = take((size_t)NENT * DFF);
    float* rel_feat = take((size_t)NREL * DM);
    float* eh       = take((size_t)NENT * HH);
    float* et       = take((size_t)NENT * HH);
    float* er       = take((size_t)NREL * HH);
    float* mbuf     = take((size_t)NENT * HH);
    float* zbuf     = take((size_t)NENT * HH);
    float* escore   = take((size_t)E * HH);

    float* out_ent = (float*)d_out;
    float* out_rel = out_ent + (size_t)NENT * DM;

    const dim3 blk(256);
    const int NH  = NENT * HH;      // 400000
    const int EH  = E * HH;         // 6400000
    const long ND = (long)NENT * DM;

    // 1) ent_feat = ent_embed @ W_ent^T
    gemm_wmma_f32_k<0><<<dim3((NENT + 127) / 128, DM / 64), blk, 0, stream>>>(
        ent_embed, W_ent, ent_feat, nullptr, nullptr, nullptr, NENT, DM, DM);
    // 2) rel_feat = rel_embed @ W_rel^T
    gemm_wmma_f32_k<0><<<dim3((NREL + 127) / 128, DM / 64), blk, 0, stream>>>(
        rel_embed, W_rel, rel_feat, nullptr, nullptr, nullptr, NREL, DM, DM);

    // 3) attention logits
    logits_kernel<<<(NH + 255) / 256, blk, 0, stream>>>(ent_feat, attn_h, eh, NENT);
    logits_kernel<<<(NH + 255) / 256, blk, 0, stream>>>(ent_feat, attn_t, et, NENT);
    logits_kernel<<<(NREL * HH + 255) / 256, blk, 0, stream>>>(rel_feat, attn_r, er, NREL);

    // 4) edge softmax over incoming edges of dst
    fill_kernel<<<1024, blk, 0, stream>>>(mbuf, -3.0e38f, NH);
    fill_kernel<<<1024, blk, 0, stream>>>(zbuf, 0.0f, NH);
    edge_score_kernel<<<(EH + 255) / 256, blk, 0, stream>>>(src, dst, lab, eh, et, er,
                                                            escore, mbuf, E);
    edge_exp_kernel<<<(EH + 255) / 256, blk, 0, stream>>>(dst, escore, mbuf, zbuf, E);
    edge_norm_kernel<<<(EH + 255) / 256, blk, 0, stream>>>(dst, escore, zbuf, E);

    // 5) two PPR hops (acc + 51MB feat buffers are L2-resident on MI455X)
    const float* fin = ent_feat;
    for (int hop = 0; hop < 2; ++hop) {
        fill_kernel<<<4096, blk, 0, stream>>>(acc, 0.0f, ND);
        hop_kernel<<<(E + 3) / 4, blk, 0, stream>>>(src, dst, escore, fin, acc, E);
        combine_kernel<<<4096, blk, 0, stream>>>((const float4*)acc, (const float4*)ent_feat,
                                                 (float4*)feat_cur, ND / 4);
        fin = feat_cur;
    }

    // 6) ent_rst = ent_embed + rw1 * (feat @ W_out^T)
    gemm_wmma_f32_k<1><<<dim3((NENT + 127) / 128, DM / 64), blk, 0, stream>>>(
        feat_cur, W_out, ent_rst, nullptr, ent_embed, rw1, NENT, DM, DM);

    // 7) rel_rst = rel_embed + rw_rel * rel_feat
    relrst_kernel<<<(NREL * DM + 255) / 256, blk, 0, stream>>>(rel_embed, rel_feat, rw_rel,
                                                               out_rel, NREL * DM);

    // 8) FFN: hid = relu(ent_rst @ w1^T + b1)
    gemm_wmma_f32_k<2><<<dim3((NENT + 127) / 128, DFF / 64), blk, 0, stream>>>(
        ent_rst, w1, hid, b1, nullptr, nullptr, NENT, DFF, DM);
    // 9) out = ent_rst + rw2 * (hid @ w2^T + b2)
    gemm_wmma_f32_k<3><<<dim3((NENT + 127) / 128, DM / 64), blk, 0, stream>>>(
        hid, w2, out_ent, b2, ent_rst, rw2, NENT, DM, DFF);
}